// SingleTransformerLayer_43164421325324
// MI455X (gfx1250) — compile-verified
//
#include <hip/hip_runtime.h>
#include <hip/hip_bf16.h>

typedef __attribute__((ext_vector_type(2))) float v2f;
typedef __attribute__((ext_vector_type(8))) float v8f;

#define BATCH 2
#define SEQ   256
#define HEADS 16
#define HD    64
#define EMBED 1024
#define ROWS  (BATCH * SEQ * HEADS)   // 8192 rows of the (rows,64) QKV GEMM
#define NS    (BATCH * SEQ)           // 512

// ---------------------------------------------------------------------------
// Kernel 1: Q/K/V projection.  Xr (8192x64) @ W^T (64x64) -> (8192x64),
// using V_WMMA_F32_16X16X4_F32. One wave computes one 16x16 tile, 16 K-steps.
// A layout: lane L -> m=L&15, VGPR pair holds K = 2*(L>>4), 2*(L>>4)+1.
// B layout: lane L -> n=L&15, same K pairing.  B[k][n] = W[n_col][k].
// ---------------------------------------------------------------------------
__global__ __launch_bounds__(32) void qkv_wmma_kernel(
    const float* __restrict__ X,
    const float* __restrict__ Wq, const float* __restrict__ Wk,
    const float* __restrict__ Wv,
    float* __restrict__ Q, float* __restrict__ Km, float* __restrict__ Vm) {
  const int rowTile = blockIdx.x;            // 0..511
  const int colTile = blockIdx.y;            // 0..3
  const int mat     = blockIdx.z;            // 0..2
  const float* W = (mat == 0) ? Wq : (mat == 1) ? Wk : Wv;
  float*       D = (mat == 0) ? Q  : (mat == 1) ? Km : Vm;

  const int lane = threadIdx.x;              // 0..31
  const int mn   = lane & 15;                // m for A, n for B
  const int pair = lane >> 4;                // K-pair select
  const int row0 = rowTile * 16;
  const int col0 = colTile * 16;

  const float* xrow = X + (size_t)(row0 + mn) * HD;  // A: row m of X tile
  const float* wrow = W + (size_t)(col0 + mn) * HD;  // B: W[col0+n][k]

  v8f c = {};
#pragma unroll
  for (int kk = 0; kk < HD; kk += 4) {
    v2f a = *(const v2f*)(xrow + kk + 2 * pair);
    v2f b = *(const v2f*)(wrow + kk + 2 * pair);
    c = __builtin_amdgcn_wmma_f32_16x16x4_f32(false, a, false, b,
                                              (short)0, c, false, false);
  }
  // C/D layout: VGPR j holds M = j + 8*pair, N = lane&15
#pragma unroll
  for (int j = 0; j < 8; ++j) {
    const int mm = j + 8 * pair;
    D[(size_t)(row0 + mm) * HD + col0 + mn] = c[j];
  }
}

// ---------------------------------------------------------------------------
// Kernel 2: head-axis softmax attention core (the reference's nonstandard op).
// Block = one (n,s); 256 threads = 64 d-lanes x 4 t-lanes.
// Per t: e[h] = (q[h]/32)*k[h]; softmax over h (16); acc[h] += p[h]*v[h].
// LDS reduce over the 4 t-lanes.
// ---------------------------------------------------------------------------
__global__ __launch_bounds__(256) void attn_kernel(
    const float* __restrict__ Q, const float* __restrict__ Km,
    const float* __restrict__ Vm, float* __restrict__ AO) {
  const int ns  = blockIdx.x;          // n*SEQ + s
  const int tid = threadIdx.x;
  const int d   = tid & 63;
  const int tl  = tid >> 6;            // 0..3
  const int nbase = (ns >> 8) << 8;    // n*SEQ
  const float scale = 0.03125f;        // 1/sqrt(EMBED) = 1/32

  float qv[16], ev[16], acc[16];
  const float* qb = Q + (size_t)ns * EMBED + d;
#pragma unroll
  for (int h = 0; h < 16; ++h) {
    qv[h] = qb[h * HD] * scale;
    acc[h] = 0.0f;
  }

  for (int t = tl; t < SEQ; t += 4) {
    const float* kb = Km + (size_t)(nbase + t) * EMBED + d;
    float mx = -3.402823466e+38f;
#pragma unroll
    for (int h = 0; h < 16; ++h) {
      float e = qv[h] * kb[h * HD];
      ev[h] = e;
      mx = fmaxf(mx, e);
    }
    float sum = 0.0f;
#pragma unroll
    for (int h = 0; h < 16; ++h) {
      float p = __expf(ev[h] - mx);
      ev[h] = p;
      sum += p;
    }
    const float inv = 1.0f / sum;
    const float* vb = Vm + (size_t)(nbase + t) * EMBED + d;
#pragma unroll
    for (int h = 0; h < 16; ++h) {
      acc[h] += ev[h] * inv * vb[h * HD];
    }
  }

  __shared__ float red[4][64][17];     // padded to dodge bank conflicts
#pragma unroll
  for (int h = 0; h < 16; ++h) red[tl][d][h] = acc[h];
  __syncthreads();

  for (int idx = tid; idx < EMBED; idx += 256) {
    const int h  = idx >> 6;
    const int dd = idx & 63;
    const float s =
        red[0][dd][h] + red[1][dd][h] + red[2][dd][h] + red[3][dd][h];
    AO[(size_t)ns * EMBED + h * HD + dd] = s;   // (n,s,h,d) -> (n,s,1024)
  }
}

// ---------------------------------------------------------------------------
// Kernel 3: output projection.  AO (512x1024) @ Wo^T (1024x1024) + bo,
// V_WMMA_F32_16X16X4_F32, one wave per 16x16 tile, 256 K-steps, fused bias.
// ---------------------------------------------------------------------------
__global__ __launch_bounds__(32) void outproj_wmma_kernel(
    const float* __restrict__ AO, const float* __restrict__ Wo,
    const float* __restrict__ bo, float* __restrict__ OUT) {
  const int rowTile = blockIdx.x;      // 0..31
  const int colTile = blockIdx.y;      // 0..63
  const int lane = threadIdx.x;
  const int mn   = lane & 15;
  const int pair = lane >> 4;
  const int row0 = rowTile * 16;
  const int col0 = colTile * 16;

  const float* arow = AO + (size_t)(row0 + mn) * EMBED;
  const float* brow = Wo + (size_t)(col0 + mn) * EMBED;  // B[k][n] = Wo[n][k]

  v8f c = {};
#pragma unroll 8
  for (int kk = 0; kk < EMBED; kk += 4) {
    v2f a = *(const v2f*)(arow + kk + 2 * pair);
    v2f b = *(const v2f*)(brow + kk + 2 * pair);
    c = __builtin_amdgcn_wmma_f32_16x16x4_f32(false, a, false, b,
                                              (short)0, c, false, false);
  }

  const float bias = bo[col0 + mn];
#pragma unroll
  for (int j = 0; j < 8; ++j) {
    const int mm = j + 8 * pair;
    OUT[(size_t)(row0 + mm) * EMBED + col0 + mn] = c[j] + bias;
  }
}

// ---------------------------------------------------------------------------
extern "C" void kernel_launch(void* const* d_in, const int* in_sizes, int n_in,
                              void* d_out, int out_size, void* d_ws,
                              size_t ws_size, hipStream_t stream) {
  const float* x  = (const float*)d_in[0];
  const float* Wq = (const float*)d_in[1];
  const float* Wk = (const float*)d_in[2];
  const float* Wv = (const float*)d_in[3];
  const float* Wo = (const float*)d_in[4];
  const float* bo = (const float*)d_in[5];
  float* out = (float*)d_out;

  const size_t QKV_ELEMS = (size_t)ROWS * HD;      // 524288 floats
  float* ws = (float*)d_ws;
  float* Q  = ws;
  float* Km = ws + QKV_ELEMS;
  float* Vm = ws + 2 * QKV_ELEMS;
  float* AO = ws + 3 * QKV_ELEMS;

  // Q/K/V projection: 512 row-tiles x 4 col-tiles x 3 matrices, 1 wave each.
  dim3 g1(ROWS / 16, HD / 16, 3);
  hipLaunchKernelGGL(qkv_wmma_kernel, g1, dim3(32), 0, stream,
                     x, Wq, Wk, Wv, Q, Km, Vm);

  // Attention core: one block per (n,s).
  hipLaunchKernelGGL(attn_kernel, dim3(NS), dim3(256), 0, stream,
                     Q, Km, Vm, AO);

  // Output projection + bias.
  dim3 g3(NS / 16, EMBED / 16);
  hipLaunchKernelGGL(outproj_wmma_kernel, g3, dim3(32), 0, stream,
                     AO, Wo, bo, out);
}